// LightGCN_38611755991225
// MI455X (gfx1250) — compile-verified
//
#include <hip/hip_runtime.h>
#include <hip/hip_bf16.h>

// ---------------- problem constants (match reference) ----------------
#define NUSERS   100000
#define NITEMS   200000
#define NNODES   300000      // NUSERS + NITEMS
#define EMB      64
#define NLAYERS  3
#define NEDGES   1000000     // interactions; 2M directed edges
#define BATCH    16384

typedef __attribute__((ext_vector_type(2))) float v2f;
typedef __attribute__((ext_vector_type(8))) float v8f;

// ---------------- small utility kernels ----------------
__global__ void lg_zero4(float4* p, int n4) {
    int t = blockIdx.x * blockDim.x + threadIdx.x;
    if (t < n4) p[t] = make_float4(0.f, 0.f, 0.f, 0.f);
}

// softmax over NLAYERS+1 = 4 layer weights, single thread
__global__ void lg_softmax(const float* lw, float* w) {
    float m = lw[0];
    for (int i = 1; i < NLAYERS + 1; ++i) m = fmaxf(m, lw[i]);
    float s = 0.f;
    float e[NLAYERS + 1];
    for (int i = 0; i < NLAYERS + 1; ++i) { e[i] = __expf(lw[i] - m); s += e[i]; }
    for (int i = 0; i < NLAYERS + 1; ++i) w[i] = e[i] / s;
}

// degree accumulation: each interaction bumps both endpoints
__global__ void lg_degree(const int* __restrict__ iu, const int* __restrict__ ii,
                          float* deg, int ne) {
    int t = blockIdx.x * blockDim.x + threadIdx.x;
    if (t < ne) {
        unsafeAtomicAdd(&deg[iu[t]], 1.0f);
        unsafeAtomicAdd(&deg[NUSERS + ii[t]], 1.0f);
    }
}

// dinv = deg^-0.5 (in place)
__global__ void lg_dinv(float* deg, int n) {
    int t = blockIdx.x * blockDim.x + threadIdx.x;
    if (t < n) {
        float d = deg[t];
        deg[t] = (d > 0.f) ? (1.0f / sqrtf(d)) : 0.f;
    }
}

// batch accumulators init with layer-0 term: ub = w0 * user_emb[uid], ib = w0 * item_emb[iid]
__global__ void lg_batch_init(const float* __restrict__ ue, const float* __restrict__ ie,
                              const int* __restrict__ uid, const int* __restrict__ iid,
                              const float* __restrict__ w,
                              float* __restrict__ ub, float* __restrict__ ib) {
    int t = blockIdx.x * blockDim.x + threadIdx.x;   // BATCH*EMB threads
    if (t < BATCH * EMB) {
        int b = t >> 6, d = t & 63;
        float w0 = w[0];
        ub[t] = w0 * ue[(size_t)uid[b] * EMB + d];
        ib[t] = w0 * ie[(size_t)iid[b] * EMB + d];
    }
}

// one wave (32 lanes) per interaction; lane handles dims 2*lane, 2*lane+1.
// out[u] += w * emb[c];  out[c] += w * emb[u];  w = dinv[u]*dinv[c]
__global__ void lg_propagate(const int* __restrict__ iu, const int* __restrict__ ii,
                             const float* __restrict__ dinv,
                             const float* __restrict__ src, float* __restrict__ dst,
                             int ne) {
    int gt   = blockIdx.x * blockDim.x + threadIdx.x;
    int edge = gt >> 5;
    int lane = gt & 31;
    if (edge >= ne) return;

    // stream-prefetch upcoming edge indices (global_prefetch_b8)
    if (lane == 0 && edge + 2048 < ne) {
        __builtin_prefetch(&iu[edge + 2048], 0, 1);
        __builtin_prefetch(&ii[edge + 2048], 0, 1);
    }

    int u = iu[edge];
    int c = NUSERS + ii[edge];
    float w = dinv[u] * dinv[c];

    const size_t ou = (size_t)u * EMB + 2 * lane;
    const size_t oc = (size_t)c * EMB + 2 * lane;
    v2f eu = *(const v2f*)(src + ou);
    v2f ec = *(const v2f*)(src + oc);

    unsafeAtomicAdd(dst + ou,     w * ec.x);
    unsafeAtomicAdd(dst + ou + 1, w * ec.y);
    unsafeAtomicAdd(dst + oc,     w * eu.x);
    unsafeAtomicAdd(dst + oc + 1, w * eu.y);
}

// accumulate weighted gathered layer embeddings into batch buffers
__global__ void lg_gather_acc(const float* __restrict__ emb,
                              const int* __restrict__ uid, const int* __restrict__ iid,
                              const float* __restrict__ w, int widx,
                              float* __restrict__ ub, float* __restrict__ ib) {
    int t = blockIdx.x * blockDim.x + threadIdx.x;
    if (t < BATCH * EMB) {
        int b = t >> 6, d = t & 63;
        float wl = w[widx];
        ub[t] += wl * emb[(size_t)uid[b] * EMB + d];
        ib[t] += wl * emb[((size_t)NUSERS + iid[b]) * EMB + d];
    }
}

// scores[b] = dot(ub[b], ib[b]) via V_WMMA_F32_16X16X4_F32.
// One wave per tile of 16 batch rows: C = U(16x64) * I^T(64x16), take diagonal.
// f32 WMMA => exact f32 RNE math, no precision loss vs scalar FMA.
__global__ void lg_score_wmma(const float* __restrict__ ub, const float* __restrict__ ib,
                              float* __restrict__ out) {
    int gt   = blockIdx.x * blockDim.x + threadIdx.x;
    int tile = gt >> 5;                 // 1024 tiles of 16
    unsigned lane = threadIdx.x & 31u;

    int m    = (int)(lane & 15u);       // row within tile (A: M, B: N)
    int koff = (int)((lane >> 4u) << 1u); // lanes 16-31 carry K+2,K+3

    const float* urow = ub + ((size_t)tile * 16 + m) * EMB;
    const float* irow = ib + ((size_t)tile * 16 + m) * EMB;

    v8f c = {};
    #pragma unroll
    for (int k0 = 0; k0 < EMB; k0 += 4) {
        v2f a = *(const v2f*)(urow + k0 + koff);   // A[m][k0+koff .. +1]
        v2f b = *(const v2f*)(irow + k0 + koff);   // B[k0+koff][n=m] = I[m][k..]
        c = __builtin_amdgcn_wmma_f32_16x16x4_f32(
                /*neg_a=*/false, a, /*neg_b=*/false, b,
                /*c_mod=*/(short)0, c, /*reuse_a=*/false, /*reuse_b=*/false);
    }

    // C/D layout: VGPR r, lanes 0-15 -> M=r,N=lane ; lanes 16-31 -> M=r+8,N=lane-16
    // diagonal t<8:  VGPR t,   lane t        ; t>=8: VGPR t-8, lane t+16
    float val = 0.f;
    int idx = -1;
    if (lane < 8u)                       { idx = (int)lane;        val = c[lane]; }
    else if (lane >= 16u && lane < 24u)  { idx = (int)(lane - 8u); val = c[lane - 16u]; }
    if (idx >= 0) out[tile * 16 + idx] = val;
}

// ---------------- launch ----------------
extern "C" void kernel_launch(void* const* d_in, const int* in_sizes, int n_in,
                              void* d_out, int out_size, void* d_ws, size_t ws_size,
                              hipStream_t stream) {
    const float* user_emb = (const float*)d_in[0];
    const float* item_emb = (const float*)d_in[1];
    const float* layer_w  = (const float*)d_in[2];
    const int*   iu       = (const int*)d_in[3];
    const int*   ii       = (const int*)d_in[4];
    const int*   uid      = (const int*)d_in[5];
    const int*   iid      = (const int*)d_in[6];
    float*       scores   = (float*)d_out;

    // workspace carve-out (256B aligned)
    char* base = (char*)d_ws;
    size_t off = 0;
    auto carve = [&](size_t bytes) -> char* {
        char* p = base + off;
        off = (off + bytes + 255) & ~(size_t)255;
        return p;
    };
    float* w    = (float*)carve((NLAYERS + 1) * sizeof(float));
    float* dinv = (float*)carve((size_t)NNODES * sizeof(float));
    float* embA = (float*)carve((size_t)NNODES * EMB * sizeof(float));
    float* embB = (float*)carve((size_t)NNODES * EMB * sizeof(float));
    float* ub   = (float*)carve((size_t)BATCH * EMB * sizeof(float));
    float* ib   = (float*)carve((size_t)BATCH * EMB * sizeof(float));
    (void)ws_size; (void)n_in; (void)in_sizes; (void)out_size;

    const int TPB = 256;

    // 1) softmax of layer weights
    lg_softmax<<<1, 1, 0, stream>>>(layer_w, w);

    // 2) degrees -> dinv
    lg_zero4<<<(NNODES / 4 + TPB - 1) / TPB, TPB, 0, stream>>>((float4*)dinv, NNODES / 4);
    lg_degree<<<(NEDGES + TPB - 1) / TPB, TPB, 0, stream>>>(iu, ii, dinv, NEDGES);
    lg_dinv<<<(NNODES + TPB - 1) / TPB, TPB, 0, stream>>>(dinv, NNODES);

    // 3) emb0 = concat(user_emb, item_emb) into embA
    hipMemcpyAsync(embA, user_emb, (size_t)NUSERS * EMB * sizeof(float),
                   hipMemcpyDeviceToDevice, stream);
    hipMemcpyAsync(embA + (size_t)NUSERS * EMB, item_emb,
                   (size_t)NITEMS * EMB * sizeof(float),
                   hipMemcpyDeviceToDevice, stream);

    // 4) batch accumulators: layer-0 contribution
    lg_batch_init<<<(BATCH * EMB + TPB - 1) / TPB, TPB, 0, stream>>>(
        user_emb, item_emb, uid, iid, w, ub, ib);

    // 5) propagation layers (ping-pong embA/embB, all resident in L2)
    float* cur = embA;
    float* nxt = embB;
    const int n4  = NNODES * EMB / 4;
    const int pgrid = (NEDGES * 32 + TPB - 1) / TPB;   // one wave32 per interaction
    for (int l = 0; l < NLAYERS; ++l) {
        lg_zero4<<<(n4 + TPB - 1) / TPB, TPB, 0, stream>>>((float4*)nxt, n4);
        lg_propagate<<<pgrid, TPB, 0, stream>>>(iu, ii, dinv, cur, nxt, NEDGES);
        lg_gather_acc<<<(BATCH * EMB + TPB - 1) / TPB, TPB, 0, stream>>>(
            nxt, uid, iid, w, l + 1, ub, ib);
        float* tmp = cur; cur = nxt; nxt = tmp;
    }

    // 6) scoring: 1024 tiles x 16 rows, 8 waves/block -> 128 blocks, EXEC all-ones per wave
    lg_score_wmma<<<(BATCH / 16) / 8, TPB, 0, stream>>>(ub, ib, scores);
}